// DistMultMod_10763188044488
// MI455X (gfx1250) — compile-verified
//
#include <hip/hip_runtime.h>

typedef __attribute__((ext_vector_type(2))) float v2f;
typedef __attribute__((ext_vector_type(8))) float v8f;

#define NNODES 500000
#define DDIM   128
#define BTRIP  8192
#define KNEI   64

// ---------------------------------------------------------------------------
// Kernel A: clear the per-node "who updated me" mark array (must be reset
// every call; harness does not re-poison/restore between graph replays).
// ---------------------------------------------------------------------------
__global__ __launch_bounds__(256) void clear_marks(int* __restrict__ mark, int n) {
  int i = blockIdx.x * 256 + threadIdx.x;
  if (i < n) mark[i] = -1;
}

// ---------------------------------------------------------------------------
// Kernel B: one wave (32 lanes) per triplet. All control data (head, local
// disease index, 64 neighbor ids, 64 weights) is wave-uniform, so we force
// it onto the scalar path with readfirstlane: neighbor ids/weights come in
// through s_load bursts (constant cache, KMcnt), and each gathered row is a
// single global_load_b128 per lane with SGPR base + constant lane offset.
// The kernel's job is purely to issue 512K random 512B row fetches at the
// HBM roofline; 8 row-loads in flight per wave x 8192 waves gives the MLP.
// Publishes:
//   upd[b][:]  = blended (or old) head row
//   mark[head] = max-b writer (deterministic duplicate resolution)
// ---------------------------------------------------------------------------
__global__ __launch_bounds__(256) void aggregate_kernel(
    const int*   __restrict__ head_index,
    const int*   __restrict__ rel_type,
    const float* __restrict__ node_emb,
    const int*   __restrict__ local_idx_map,
    const int*   __restrict__ sim_neighbors,
    const float* __restrict__ sim_weights,
    const int*   __restrict__ degree_table,
    float*       __restrict__ upd,
    int*         __restrict__ mark) {
  const int lane = threadIdx.x & 31;
  // wave id is uniform by construction; make it provably scalar.
  const int b = __builtin_amdgcn_readfirstlane(
      (int)((blockIdx.x * 256 + threadIdx.x) >> 5));

  const int head = head_index[b];        // s_load (b scalar)
  const int rt   = rel_type[b];          // s_load
  const bool masked = (rt >= 2) && (rt <= 4);

  const float4 oldv = ((const float4*)(node_emb + (size_t)head * DDIM))[lane];
  float4 res = oldv;

  if (masked) {                          // uniform branch -> s_cbranch
    const int   local = local_idx_map[head];                 // s_load
    const int   rl    = rt - 2;                              // 0..2
    const float deg   = (float)degree_table[local * 3 + rl]; // s_load
    const float c     = 0.7f * __expf(-0.7f * deg) + 0.2f;

    const int*   np = sim_neighbors + (size_t)local * KNEI;  // 256B-aligned row
    const float* wp = sim_weights   + (size_t)local * KNEI;

    float4 acc = {0.f, 0.f, 0.f, 0.f};
    for (int k0 = 0; k0 < KNEI; k0 += 8) {                   // 8 loads in flight
      float4 v[8];
#pragma unroll
      for (int j = 0; j < 8; ++j) {
        const int n = np[k0 + j];        // scalar load (merged s_load bursts)
        v[j] = ((const float4*)(node_emb + (size_t)n * DDIM))[lane];
      }
#pragma unroll
      for (int j = 0; j < 8; ++j) {
        const float wk = wp[k0 + j];     // scalar load -> SGPR fma operand
        acc.x += wk * v[j].x;  acc.y += wk * v[j].y;
        acc.z += wk * v[j].z;  acc.w += wk * v[j].w;
      }
    }
    const float omc = 1.0f - c;
    res.x = c * acc.x + omc * oldv.x;  res.y = c * acc.y + omc * oldv.y;
    res.z = c * acc.z + omc * oldv.z;  res.w = c * acc.w + omc * oldv.w;
  }

  ((float4*)(upd + (size_t)b * DDIM))[lane] = res;
  if (lane == 0) atomicMax(&mark[head], b);
}

// ---------------------------------------------------------------------------
// Kernel C: DistMult scores via WMMA. 16 triplets per wave:
//   p = h.*r  (h from upd via mark; heads are always marked)
//   t resolved through mark (collision with an updated head -> upd row)
//   score = diag( P(16x128) . T^T(128x16) ), K=128 chained as 32x
//   V_WMMA_F32_16X16X4_F32 accumulations (f32, matches reference numerics).
// A-frag (16x4 f32): lane l (m=l&15, half=l>>4) holds K = 4*kc+2*half, +1.
// B-frag mirrored with n=l&15. C/D diag (i,i): i<8 -> lane i, VGPR i;
// i>=8 -> lane i+16, VGPR i-8.
// No divergence around the WMMAs: every block is full, EXEC all ones.
// ---------------------------------------------------------------------------
__global__ __launch_bounds__(128) void score_kernel(
    const int*   __restrict__ head_index,
    const int*   __restrict__ rel_type,
    const int*   __restrict__ tail_index,
    const float* __restrict__ node_emb,
    const float* __restrict__ rel_emb,
    const float* __restrict__ upd,
    const int*   __restrict__ mark,
    float*       __restrict__ out) {
  const int lane  = threadIdx.x & 31;
  const int wave  = blockIdx.x * 4 + (threadIdx.x >> 5);
  const int base  = wave * 16;               // 16 triplets per wave
  const int m     = lane & 15;
  const int half  = lane >> 4;
  const int b     = base + m;

  const int head = head_index[b];
  const int tail = tail_index[b];
  const int rt   = rel_type[b];

  const float* hrow = upd + (size_t)mark[head] * DDIM;       // heads always marked
  const int    tm   = mark[tail];
  const float* trow = (tm >= 0) ? (upd + (size_t)tm * DDIM)
                                : (node_emb + (size_t)tail * DDIM);
  const float* rrow = rel_emb + (size_t)rt * DDIM;

  v8f c = {0.f, 0.f, 0.f, 0.f, 0.f, 0.f, 0.f, 0.f};
#pragma unroll 4
  for (int kc = 0; kc < 32; ++kc) {
    const int off = 4 * kc + 2 * half;
    const float2 hv = *(const float2*)(hrow + off);
    const float2 rv = *(const float2*)(rrow + off);
    const float2 tv = *(const float2*)(trow + off);
    v2f a  = {hv.x * rv.x, hv.y * rv.y};
    v2f bb = {tv.x, tv.y};
    c = __builtin_amdgcn_wmma_f32_16x16x4_f32(
        /*neg_a=*/false, a, /*neg_b=*/false, bb,
        /*c_mod=*/(short)0, c, /*reuse_a=*/false, /*reuse_b=*/false);
  }

  // Extract diagonal element for this lane's triplet (select chain avoids
  // dynamic vector subscript).
  const int idx = (lane < 8) ? lane : (lane - 24);           // VGPR index
  float diag = 0.f;
#pragma unroll
  for (int j = 0; j < 8; ++j) diag = (idx == j) ? c[j] : diag;

  if (lane < 8)        out[base + lane]        = diag;       // i = 0..7
  else if (lane >= 24) out[base + (lane - 16)] = diag;       // i = 8..15
}

// ---------------------------------------------------------------------------
extern "C" void kernel_launch(void* const* d_in, const int* in_sizes, int n_in,
                              void* d_out, int out_size, void* d_ws, size_t ws_size,
                              hipStream_t stream) {
  const int*   head_index    = (const int*)  d_in[0];
  const int*   rel_type      = (const int*)  d_in[1];
  const int*   tail_index    = (const int*)  d_in[2];
  const float* node_emb      = (const float*)d_in[3];
  const float* rel_emb       = (const float*)d_in[4];
  const int*   local_idx_map = (const int*)  d_in[5];
  const int*   sim_neighbors = (const int*)  d_in[6];
  const float* sim_weights   = (const float*)d_in[7];
  const int*   degree_table  = (const int*)  d_in[8];
  float*       out           = (float*)d_out;

  // Workspace layout: [0, 2MB) mark[N] ints; [2MB, 6MB) upd[B][128] floats.
  int*   mark = (int*)d_ws;
  float* upd  = (float*)((char*)d_ws + (size_t)(2u << 20));

  clear_marks<<<(NNODES + 255) / 256, 256, 0, stream>>>(mark, NNODES);

  aggregate_kernel<<<BTRIP / 8, 256, 0, stream>>>(
      head_index, rel_type, node_emb, local_idx_map,
      sim_neighbors, sim_weights, degree_table, upd, mark);

  score_kernel<<<BTRIP / 64, 128, 0, stream>>>(
      head_index, rel_type, tail_index, node_emb, rel_emb, upd, mark, out);
}